// MixedLayer_25340307047010
// MI455X (gfx1250) — compile-verified
//
#include <hip/hip_runtime.h>
#include <hip/hip_bf16.h>

typedef __attribute__((ext_vector_type(2))) float v2f;
typedef __attribute__((ext_vector_type(8))) float v8f;

#define F_   32
#define OPS_ 5
#define CIN_ 32
#define NB_  16
#define H_   128
#define W_   128
#define KTOT 288   // 9 taps * 32 channels

// --- CDNA5 async global->LDS path (bypasses VGPRs, tracked by ASYNCcnt) ----
#if defined(__gfx1250__) && __has_builtin(__builtin_amdgcn_global_load_async_to_lds_b32)
  typedef __attribute__((address_space(1))) int* gas1_i32_t;   // "__device__ int*"
  typedef __attribute__((address_space(3))) int* las3_i32_t;   // "__shared__ int*"
  #define ASYNC_LDS_LOAD_B32(g, l) \
      __builtin_amdgcn_global_load_async_to_lds_b32((gas1_i32_t)(g), \
                                                    (las3_i32_t)(l), 0, 0)
  #if __has_builtin(__builtin_amdgcn_s_wait_asynccnt)
    #define ASYNC_WAIT() __builtin_amdgcn_s_wait_asynccnt(0)
  #else
    #define ASYNC_WAIT() asm volatile("s_wait_asynccnt 0x0" ::: "memory")
  #endif
  #define HAVE_ASYNC 1
#else
  #define HAVE_ASYNC 0
#endif

// ---------------------------------------------------------------------------
// Kernel 0: zero the stat accumulators (ws[0..63]) every launch (determinism).
// ---------------------------------------------------------------------------
__global__ void zero_stats_kernel(float* __restrict__ ws) {
    if (threadIdx.x < 64) ws[threadIdx.x] = 0.0f;
}

// ---------------------------------------------------------------------------
// Kernel 1: implicit-GEMM 3x3 conv via V_WMMA_F32_16X16X4_F32.
// Block = 256 threads (8 waves) handles one (n, h) output row; wave w covers
// output columns [16w, 16w+16) for all 32 filters (two 16x16 M-tiles).
// K order: k = tap*32 + c (tap = kh*3+kw). Halo rows stream into LDS via
// GLOBAL_LOAD_ASYNC_TO_LDS_B32 when available.
// Also accumulates per-filter sum / sum-of-squares for BatchNorm.
// ---------------------------------------------------------------------------
__global__ __launch_bounds__(256) void conv_wmma_kernel(
    const float* __restrict__ x,       // [N, CIN, H, W]
    const float* __restrict__ W_all,   // [F, OPS, CIN, 3, 3]
    const int*   __restrict__ op_idx,  // [F]
    float* __restrict__ out1,          // [N, F, H, W]  (conv result, pre-BN)
    float* __restrict__ gsum,          // [32]
    float* __restrict__ gsq)           // [32]
{
    __shared__ float Wlds[F_][KTOT];       // 36,864 B  gathered weights [f][k]
    __shared__ float xlds[3][16][132];     // 25,344 B  3 halo rows x 16 ch x (pad+128+pad)
    __shared__ float bsum[F_];
    __shared__ float bsq[F_];

    const int tid  = threadIdx.x;
    const int lane = tid & 31;
    const int wave = tid >> 5;
    const int n = blockIdx.x >> 7;     // blockIdx.x / 128
    const int h = blockIdx.x & 127;

    // ---- stage gathered weights: Wlds[f][tap*32+c] = W_all[f, op_idx[f], c, kh, kw]
    for (int idx = tid; idx < F_ * KTOT; idx += 256) {
        const int f = idx / KTOT;
        const int k = idx - f * KTOT;
        const int tap = k >> 5;        // k / 32
        const int c   = k & 31;
        const int op  = op_idx[f];
        // W_all element (f,op,c,kh,kw) at (f*OPS+op)*288 + c*9 + tap
        Wlds[f][k] = W_all[(f * OPS_ + op) * 288 + c * 9 + tap];
    }
    if (tid < F_) { bsum[tid] = 0.0f; bsq[tid] = 0.0f; }
    // ---- pre-zero SAME-padding columns (cc=0 -> w=-1, cc=129 -> w=128);
    //      staging below only writes cc=1..128, so pads stay zero for both chunks.
    for (int idx = tid; idx < 3 * 16; idx += 256) {
        const int r = idx >> 4, cl = idx & 15;
        xlds[r][cl][0]   = 0.0f;
        xlds[r][cl][129] = 0.0f;
    }

    v8f acc0 = {0.f,0.f,0.f,0.f,0.f,0.f,0.f,0.f};
    v8f acc1 = {0.f,0.f,0.f,0.f,0.f,0.f,0.f,0.f};

    const int khalf = (lane >> 4) << 1;       // lanes 0-15: K pair {k0,k0+1}; 16-31: {k0+2,k0+3}
    const int fr    = lane & 15;              // A-matrix row (filter) within M-tile
    const int col   = (wave << 4) | (lane & 15);  // output column 0..127

    for (int cblk = 0; cblk < 2; ++cblk) {
        __syncthreads();   // previous chunk fully consumed (and pads written) before reuse
#if HAVE_ASYNC
        // 48 (row, channel) halo lines split across the 8 waves; each valid line
        // streams 128 floats straight into LDS via the async data mover path.
        for (int p = wave; p < 48; p += 8) {
            const int r  = p / 16;
            const int cl = p - r * 16;
            const int hh = h + r - 1;
            if (hh >= 0 && hh < H_) {
                const float* gsrc = x + ((n * CIN_ + (cblk * 16 + cl)) * H_ + hh) * W_;
                #pragma unroll
                for (int j = 0; j < 4; ++j) {
                    const int w = j * 32 + lane;
                    ASYNC_LDS_LOAD_B32(gsrc + w, &xlds[r][cl][1 + w]);
                }
            } else {
                #pragma unroll
                for (int j = 0; j < 4; ++j)
                    xlds[r][cl][1 + j * 32 + lane] = 0.0f;
            }
        }
        ASYNC_WAIT();
#else
        for (int idx = tid; idx < 3 * 16 * 128; idx += 256) {
            const int r   = idx / (16 * 128);
            const int rem = idx - r * (16 * 128);
            const int cl  = rem >> 7;
            const int w   = rem & 127;
            const int hh  = h + r - 1;
            float v = 0.0f;
            if (hh >= 0 && hh < H_)
                v = x[((n * CIN_ + (cblk * 16 + cl)) * H_ + hh) * W_ + w];
            xlds[r][cl][1 + w] = v;
        }
#endif
        __syncthreads();

        for (int tap = 0; tap < 9; ++tap) {
            const int kh = tap / 3;
            const int kw = tap - kh * 3;
            #pragma unroll
            for (int cl = 0; cl < 16; cl += 4) {
                const int k0 = tap * 32 + cblk * 16 + cl;
                // A operands: lane fr holds Wsel[fr (+16)][k0+khalf, k0+khalf+1] (8B-aligned)
                v2f a0 = *(const v2f*)&Wlds[fr][k0 + khalf];
                v2f a1 = *(const v2f*)&Wlds[fr + 16][k0 + khalf];
                // B operand: x[n, c+khalf(+1), h+kh-1, col+kw-1] from halo (index col+kw)
                v2f b;
                b.x = xlds[kh][cl + khalf][col + kw];
                b.y = xlds[kh][cl + khalf + 1][col + kw];
                acc0 = __builtin_amdgcn_wmma_f32_16x16x4_f32(
                    false, a0, false, b, (short)0, acc0, false, false);
                acc1 = __builtin_amdgcn_wmma_f32_16x16x4_f32(
                    false, a1, false, b, (short)0, acc1, false, false);
            }
        }
    }

    // ---- write conv output + per-filter batch statistics
    // C/D layout: VGPR i, lanes0-15 -> M=i ; lanes16-31 -> M=i+8 ; N = lane&15.
    const int hi8 = (lane >> 4) << 3;
    #pragma unroll
    for (int i = 0; i < 8; ++i) {
        const int f0 = i + hi8;
        const int f1 = f0 + 16;
        const float v0 = acc0[i];
        const float v1 = acc1[i];
        out1[((n * F_ + f0) * H_ + h) * W_ + col] = v0;
        out1[((n * F_ + f1) * H_ + h) * W_ + col] = v1;
        atomicAdd(&bsum[f0], v0);
        atomicAdd(&bsq[f0],  v0 * v0);
        atomicAdd(&bsum[f1], v1);
        atomicAdd(&bsq[f1],  v1 * v1);
    }
    __syncthreads();
    if (tid < F_) {
        unsafeAtomicAdd(&gsum[tid], bsum[tid]);
        unsafeAtomicAdd(&gsq[tid],  bsq[tid]);
    }
}

// ---------------------------------------------------------------------------
// Kernel 2: fold sums into per-channel affine (scale, bias).
// ---------------------------------------------------------------------------
__global__ void stats_kernel(const float* __restrict__ gsum,
                             const float* __restrict__ gsq,
                             const float* __restrict__ gamma,
                             const float* __restrict__ beta,
                             float* __restrict__ sb)   // sb[0..31]=scale, sb[32..63]=bias
{
    const int f = threadIdx.x;
    if (f < F_) {
        const float inv = 1.0f / (float)(NB_ * H_ * W_);
        const float mean = gsum[f] * inv;
        const float var  = gsq[f] * inv - mean * mean;
        const float rstd = rsqrtf(var + 1e-5f);
        const float sc   = gamma[f] * rstd;
        sb[f]       = sc;
        sb[32 + f]  = beta[f] - mean * sc;
    }
}

// ---------------------------------------------------------------------------
// Kernel 3: in-place BN-affine + ReLU, float4 vectorized. 8,388,608 elems.
// Channel of float4 index i: plane = i>>12 (4096 float4 per HxW plane), f = plane & 31.
// ---------------------------------------------------------------------------
__global__ __launch_bounds__(256) void bn_relu_kernel(float* __restrict__ out,
                                                      const float* __restrict__ sb)
{
    const int i = blockIdx.x * 256 + threadIdx.x;   // float4 index, grid sized exactly
    const int f = (i >> 12) & 31;
    const float sc = sb[f];
    const float bi = sb[32 + f];
    float4 v = ((const float4*)out)[i];
    v.x = fmaxf(fmaf(v.x, sc, bi), 0.0f);
    v.y = fmaxf(fmaf(v.y, sc, bi), 0.0f);
    v.z = fmaxf(fmaf(v.z, sc, bi), 0.0f);
    v.w = fmaxf(fmaf(v.w, sc, bi), 0.0f);
    ((float4*)out)[i] = v;
}

// ---------------------------------------------------------------------------
extern "C" void kernel_launch(void* const* d_in, const int* in_sizes, int n_in,
                              void* d_out, int out_size, void* d_ws, size_t ws_size,
                              hipStream_t stream) {
    const float* x      = (const float*)d_in[0];
    const float* W_all  = (const float*)d_in[1];
    const float* gamma  = (const float*)d_in[2];
    const float* beta   = (const float*)d_in[3];
    const int*   op_idx = (const int*)d_in[4];
    float* out = (float*)d_out;
    float* ws  = (float*)d_ws;   // [0..31]=sum [32..63]=sumsq [64..127]=scale/bias

    zero_stats_kernel<<<1, 64, 0, stream>>>(ws);
    conv_wmma_kernel<<<NB_ * H_, 256, 0, stream>>>(x, W_all, op_idx, out, ws, ws + 32);
    stats_kernel<<<1, 32, 0, stream>>>(ws, ws + 32, gamma, beta, ws + 64);
    bn_relu_kernel<<<(NB_ * F_ * H_ * W_) / 4 / 256, 256, 0, stream>>>(out, ws + 64);
}